// GraphRandomNeuralFeatures_46445776339566
// MI455X (gfx1250) — compile-verified
//
#include <hip/hip_runtime.h>
#include <hip/hip_bf16.h>

// ---------------------------------------------------------------------------
// Sizes (fixed by the reference)
// ---------------------------------------------------------------------------
#define BB   8
#define NN   64
#define FF   8
#define MM1  154
#define MM2  358
#define HH   8
#define DD   9
#define KF   48            // padded feature length (45 features + 1 const + 2 pad)
#define NC1  (MM1*HH)      // 1232 = 77*16
#define NC2  (MM2*HH)      // 2864 = 179*16
#define SCALE_OUT 0.044194173824159216f  // 1/sqrt(512)

typedef __attribute__((ext_vector_type(2))) float v2f;
typedef __attribute__((ext_vector_type(8))) float v8f;

// ---------------------------------------------------------------------------
// Kernel 0: build combined weight matrices in workspace.
//   W1c  [48][NC1] : T1  = F * W1c   (p order dg,rs,cs,ts,ds; const row = b_eq1)
//   WRow [48][NC2] : Row' (dg<-p9, rs<-p3, cs<-p5, ts<-p11, ds<-p13, const<-b_all)
//   WCol [48][NC2] : Col' (dg<-p10, rs<-p4, cs<-p6)
//   WDiag[48][NC2] : Diag'(dg[d<8]<-p0+p1+p2, dg[8]<-p2, rs<-p7, cs<-p8,
//                          ts<-p12, ds<-p14, const<-b_diag)
// ---------------------------------------------------------------------------
__device__ __forceinline__ float w2at(const float* w, int m, int p, int d, int h) {
    return w[(((m * 15) + p) * 9 + d) * 8 + h];
}

__global__ void prep_weights(const float* __restrict__ w_eq1,
                             const float* __restrict__ b_eq1,
                             const float* __restrict__ w_eq2,
                             const float* __restrict__ b_all,
                             const float* __restrict__ b_diag,
                             float* __restrict__ W1c,
                             float* __restrict__ WRow,
                             float* __restrict__ WCol,
                             float* __restrict__ WDiag) {
    const int n1 = KF * NC1;
    const int n2 = KF * NC2;
    int idx = blockIdx.x * blockDim.x + threadIdx.x;
    if (idx < n1) {
        int k = idx / NC1, mh = idx % NC1, m = mh >> 3, h = mh & 7;
        float v = 0.f;
        if (k < 45) { int p = k / 9, d = k % 9; v = w_eq1[(((m * 5) + p) * 9 + d) * 8 + h]; }
        else if (k == 45) v = b_eq1[m * 8 + h];
        W1c[idx] = v;
        return;
    }
    idx -= n1;
    if (idx >= 3 * n2) return;
    int mat = idx / n2, r = idx % n2;
    int k = r / NC2, mh = r % NC2, m = mh >> 3, h = mh & 7;
    float v = 0.f;
    int slot = (k < 45) ? (k / 9) : -1;   // 0 dg, 1 rs, 2 cs, 3 ts, 4 ds
    int d = (k < 45) ? (k % 9) : 0;
    if (mat == 0) {            // WRow
        const int pmap[5] = {9, 3, 5, 11, 13};
        if (slot >= 0)      v = w2at(w_eq2, m, pmap[slot], d, h);
        else if (k == 45)   v = b_all[m * 8 + h];
    } else if (mat == 1) {     // WCol
        const int pmap[5] = {10, 4, 6, -1, -1};
        if (slot >= 0 && pmap[slot] >= 0) v = w2at(w_eq2, m, pmap[slot], d, h);
    } else {                   // WDiag
        if (slot == 0) {
            v = w2at(w_eq2, m, 2, d, h);
            if (d < 8) v += w2at(w_eq2, m, 0, d, h) + w2at(w_eq2, m, 1, d, h);
        } else if (slot == 1) v = w2at(w_eq2, m, 7, d, h);
        else if (slot == 2)   v = w2at(w_eq2, m, 8, d, h);
        else if (slot == 3)   v = w2at(w_eq2, m, 12, d, h);
        else if (slot == 4)   v = w2at(w_eq2, m, 14, d, h);
        else if (k == 45)     v = b_diag[m * 8 + h];
    }
    (mat == 0 ? WRow : (mat == 1 ? WCol : WDiag))[r] = v;
}

// ---------------------------------------------------------------------------
// Kernel 1: node features  F[(b*64+n)][48]
//   [0..8]=dg  [9..17]=rs  [18..26]=cs  [27..35]=ts  [36..44]=ds  [45]=1  pad=0
// ---------------------------------------------------------------------------
__global__ __launch_bounds__(64) void node_features(const float* __restrict__ X,
                                                    const float* __restrict__ A,
                                                    float* __restrict__ Fo) {
    const int b = blockIdx.x, t = threadIdx.x;
    const float* Ab = A + b * NN * NN;
    const float* Xb = X + b * NN * FF;
    __shared__ float red1[64], red2[64], sx[8];
    float rsu = 0.f, csu = 0.f;
    for (int j = 0; j < NN; ++j) rsu += Ab[t * NN + j];
    for (int i = 0; i < NN; ++i) csu += Ab[i * NN + t];
    red1[t] = Ab[t * 65];   // diagonal element
    red2[t] = rsu;
    __syncthreads();
    for (int s = 32; s > 0; s >>= 1) {
        if (t < s) { red1[t] += red1[t + s]; red2[t] += red2[t + s]; }
        __syncthreads();
    }
    if (t < 8) { float s = 0.f; for (int n = 0; n < NN; ++n) s += Xb[n * FF + t]; sx[t] = s; }
    __syncthreads();
    const float trace = red1[0], sumA = red2[0];
    const float inv64 = 1.f / 64.f, inv4096 = 1.f / 4096.f;
    float* Fr = Fo + (b * NN + t) * KF;
    for (int f = 0; f < 8; ++f) {
        float x = Xb[t * FF + f];
        Fr[f]      = x;
        Fr[9 + f]  = x * inv64;
        Fr[18 + f] = x * inv64;
        Fr[27 + f] = sx[f] * inv4096;
        Fr[36 + f] = sx[f] * inv64;
    }
    Fr[8]  = Ab[t * 65];
    Fr[17] = rsu * inv64;
    Fr[26] = csu * inv64;
    Fr[35] = sumA * inv4096;
    Fr[44] = trace * inv64;
    Fr[45] = 1.f; Fr[46] = 0.f; Fr[47] = 0.f;
}

// ---------------------------------------------------------------------------
// Kernel 2: f32 WMMA GEMM  OUT[512][ncols] = F[512][48] * W[48][ncols]
// One wave per 16x16 tile; 12 chained V_WMMA_F32_16X16X4_F32.
// A frag (16x4): lanes 0-15 -> K0,K1 ; lanes 16-31 -> K2,K3  (ISA layout)
// C/D (16x16):  VGPR v -> rows v (lanes 0-15) and v+8 (lanes 16-31)
// ---------------------------------------------------------------------------
__global__ __launch_bounds__(32) void gemm_wmma_f32(const float* __restrict__ Fm,
                                                    const float* __restrict__ W,
                                                    float* __restrict__ OUT,
                                                    int ncols) {
    const int lane = threadIdx.x;
    const int half = lane >> 4;        // 0: K0/K1, 1: K2/K3
    const int l    = lane & 15;
    const int c0 = blockIdx.x * 16;    // output column tile
    const int r0 = blockIdx.y * 16;    // output row tile
    v8f acc = {};
    for (int kb = 0; kb < KF; kb += 4) {
        const float* fr = Fm + (r0 + l) * KF + kb + half * 2;
        v2f a; a.x = fr[0]; a.y = fr[1];
        const float* wr = W + (kb + half * 2) * ncols + c0 + l;
        v2f bf; bf.x = wr[0]; bf.y = wr[ncols];
        acc = __builtin_amdgcn_wmma_f32_16x16x4_f32(
            /*neg_a=*/false, a, /*neg_b=*/false, bf,
            /*c_mod=*/(short)0, acc, /*reuse_a=*/false, /*reuse_b=*/false);
    }
    float* o = OUT + (r0 + half * 8) * (size_t)ncols + c0 + l;
#pragma unroll
    for (int v = 0; v < 8; ++v) o[v * (size_t)ncols] = acc[v];
}

// ---------------------------------------------------------------------------
// Kernel 3: psi1.  out[b*512+m] = (sum_{n,h} relu(T1)*w_in1 /64 - psi0_1)/sqrt(512)
// ---------------------------------------------------------------------------
__global__ __launch_bounds__(64) void psi1_kernel(const float* __restrict__ T1,
                                                  const float* __restrict__ b_eq1,
                                                  const float* __restrict__ w_in1,
                                                  float* __restrict__ out) {
    const int bm = blockIdx.x, b = bm / MM1, m = bm % MM1, n = threadIdx.x;
    const float* t = T1 + (size_t)(b * NN + n) * NC1 + m * 8;
    float acc = 0.f;
#pragma unroll
    for (int h = 0; h < 8; ++h) acc += fmaxf(t[h], 0.f) * w_in1[m * 8 + h];
    __shared__ float red[64];
    red[n] = acc; __syncthreads();
    for (int s = 32; s > 0; s >>= 1) { if (n < s) red[n] += red[n + s]; __syncthreads(); }
    if (n == 0) {
        float psi0 = 0.f;
        for (int h = 0; h < 8; ++h) psi0 += fmaxf(b_eq1[m * 8 + h], 0.f) * w_in1[m * 8 + h];
        out[b * 512 + m] = (red[0] * (1.f / 64.f) - psi0) * SCALE_OUT;
    }
}

// ---------------------------------------------------------------------------
// Kernel 4: psi2 (heavy elementwise). One block per (b,m).
// T2 = Row'[i] + Col'[j] + A_ij*u0 + A_ji*u1 (+ Diag'[i] if i==j); relu; reduce.
// ---------------------------------------------------------------------------
__global__ __launch_bounds__(256) void psi2_kernel(const float* __restrict__ A,
                                                   const float* __restrict__ RowO,
                                                   const float* __restrict__ ColO,
                                                   const float* __restrict__ DiagO,
                                                   const float* __restrict__ w_eq2,
                                                   const float* __restrict__ b_all,
                                                   const float* __restrict__ b_diag,
                                                   const float* __restrict__ w_in2,
                                                   float* __restrict__ out) {
    const int bm = blockIdx.x, b = bm / MM2, m = bm % MM2, t = threadIdx.x;
    __shared__ float As[NN * NN];
    __shared__ float RowL[NN * HH], ColL[NN * HH], DiagL[NN * HH];
    __shared__ float r0s[8], rDs[8];
    const float* Ab = A + b * NN * NN;
    for (int s = t; s < NN * NN; s += 256) As[s] = Ab[s];
    for (int s = t; s < NN * HH; s += 256) {
        int i = s >> 3, h = s & 7;
        size_t off = (size_t)(b * NN + i) * NC2 + m * 8 + h;
        RowL[s]  = RowO[off];
        ColL[s]  = ColO[off];
        DiagL[s] = DiagO[off];
    }
    float u0[8], u1[8], w0[8], w1[8];
#pragma unroll
    for (int h = 0; h < 8; ++h) {
        u0[h] = w2at(w_eq2, m, 0, 8, h);
        u1[h] = w2at(w_eq2, m, 1, 8, h);
        w0[h] = w_in2[m * 16 + h];
        w1[h] = w_in2[m * 16 + 8 + h];
    }
    __syncthreads();
    float acc0 = 0.f, accD = 0.f;
    for (int s = 0; s < 16; ++s) {
        int c = s * 256 + t;
        int i = c >> 6, j = c & 63;
        float aij = As[c];
        float aji = As[j * 64 + i];
        bool dg = (i == j);
#pragma unroll
        for (int h = 0; h < 8; ++h) {
            float v = RowL[i * 8 + h] + ColL[j * 8 + h] + aij * u0[h] + aji * u1[h];
            if (dg) v += DiagL[i * 8 + h];
            v = fmaxf(v, 0.f);
            acc0 += v * w0[h];
            if (dg) accD += v * w1[h];
        }
    }
    // wave32 reduction then cross-wave via LDS
    for (int off = 16; off > 0; off >>= 1) {
        acc0 += __shfl_down(acc0, off);
        accD += __shfl_down(accD, off);
    }
    if ((t & 31) == 0) { r0s[t >> 5] = acc0; rDs[t >> 5] = accD; }
    __syncthreads();
    if (t == 0) {
        float a0 = 0.f, aD = 0.f;
        for (int w = 0; w < 8; ++w) { a0 += r0s[w]; aD += rDs[w]; }
        float psi0 = 0.f;
        for (int h = 0; h < 8; ++h) {
            float bb = b_all[m * 8 + h] + b_diag[m * 8 + h];
            psi0 += fmaxf(bb, 0.f) * (w_in2[m * 16 + h] + w_in2[m * 16 + 8 + h]);
        }
        out[b * 512 + MM1 + m] = (a0 * (1.f / 4096.f) + aD * (1.f / 64.f) - psi0) * SCALE_OUT;
    }
}

// ---------------------------------------------------------------------------
// Launch
// ---------------------------------------------------------------------------
extern "C" void kernel_launch(void* const* d_in, const int* in_sizes, int n_in,
                              void* d_out, int out_size, void* d_ws, size_t ws_size,
                              hipStream_t stream) {
    const float* X       = (const float*)d_in[0];
    const float* A       = (const float*)d_in[1];
    const float* w_eq1   = (const float*)d_in[2];
    const float* b_eq1   = (const float*)d_in[3];
    const float* w_in1   = (const float*)d_in[4];
    const float* w_eq2   = (const float*)d_in[6];
    const float* b_all   = (const float*)d_in[7];
    const float* b_diag  = (const float*)d_in[8];
    const float* w_in2   = (const float*)d_in[9];
    float* out = (float*)d_out;
    float* ws = (float*)d_ws;

    // workspace layout (floats)
    float* Fm    = ws;                          // 512*48      = 24576
    float* W1c   = Fm    + 512 * KF;            // 48*1232     = 59136
    float* WRow  = W1c   + KF * NC1;            // 48*2864     = 137472
    float* WCol  = WRow  + KF * NC2;
    float* WDiag = WCol  + KF * NC2;
    float* T1o   = WDiag + KF * NC2;            // 512*1232    = 630784
    float* RowO  = T1o   + 512 * NC1;           // 512*2864    = 1466368
    float* ColO  = RowO  + 512 * NC2;
    float* DiagO = ColO  + 512 * NC2;           // total ~21 MB

    const int prepN = KF * NC1 + 3 * KF * NC2;
    prep_weights<<<(prepN + 255) / 256, 256, 0, stream>>>(
        w_eq1, b_eq1, w_eq2, b_all, b_diag, W1c, WRow, WCol, WDiag);

    node_features<<<BB, 64, 0, stream>>>(X, A, Fm);

    gemm_wmma_f32<<<dim3(NC1 / 16, 512 / 16), 32, 0, stream>>>(Fm, W1c,   T1o,  NC1);
    gemm_wmma_f32<<<dim3(NC2 / 16, 512 / 16), 32, 0, stream>>>(Fm, WRow,  RowO, NC2);
    gemm_wmma_f32<<<dim3(NC2 / 16, 512 / 16), 32, 0, stream>>>(Fm, WCol,  ColO, NC2);
    gemm_wmma_f32<<<dim3(NC2 / 16, 512 / 16), 32, 0, stream>>>(Fm, WDiag, DiagO, NC2);

    psi1_kernel<<<BB * MM1, 64, 0, stream>>>(T1o, b_eq1, w_in1, out);
    psi2_kernel<<<BB * MM2, 256, 0, stream>>>(A, RowO, ColO, DiagO,
                                              w_eq2, b_all, b_diag, w_in2, out);
}